// dotproductattention_70540542869845
// MI455X (gfx1250) — compile-verified
//
#include <hip/hip_runtime.h>

typedef float v2f __attribute__((ext_vector_type(2)));
typedef float v4f __attribute__((ext_vector_type(4)));
typedef float v8f __attribute__((ext_vector_type(8)));

#define BATCH 2
#define SEQL  512
#define DIN   256
#define ODIM  256

// ---------------------------------------------------------------------------
// Projection GEMM via V_WMMA_F32_16X16X4_F32.
// out[m,n] = sum_k X[m,k]*W[k,n] + bias[n], M = BATCH*SEQL = 1024.
// Block = 256 threads = 8 waves. Block owns a 16(M) x 256(N) strip; each wave
// owns two 16x16 N-tiles (n0 = wave*32, wave*32+16), looping K in steps of 4.
//
// fp32 16x16x4 fragment layout (ISA 7.12.2):
//   A (16x4):  lane l -> row m=l&15; vgpr r -> k = k0 + 2*(l>>4) + r
//   B (4x16):  lane l -> col n=l&15 (+half select); vgpr r -> k = k0 + 2*(l>>4) + r
//   C/D (16x16): lane l, vgpr r -> D[m = r + 8*(l>>4), n = l&15]
// ---------------------------------------------------------------------------
__global__ __launch_bounds__(256) void proj_wmma_kernel(
    const float* __restrict__ X, const float* __restrict__ W,
    const float* __restrict__ bias, float* __restrict__ out) {
  const int wave = threadIdx.x >> 5;
  const int lane = threadIdx.x & 31;
  const int half = lane >> 4;
  const int lr   = lane & 15;
  const int m0   = blockIdx.x * 16;
  const int n0   = wave * 32;

  v8f c0 = {};
  v8f c1 = {};
  for (int k0 = 0; k0 < DIN; k0 += 4) {
    const int ka = k0 + 2 * half;
    v2f a;
    a.x = X[(size_t)(m0 + lr) * DIN + ka];        // contiguous pair -> b64 load
    a.y = X[(size_t)(m0 + lr) * DIN + ka + 1];
    v2f b0, b1;
    b0.x = W[(size_t)ka       * ODIM + n0 + lr];
    b0.y = W[(size_t)(ka + 1) * ODIM + n0 + lr];
    b1.x = W[(size_t)ka       * ODIM + n0 + 16 + lr];
    b1.y = W[(size_t)(ka + 1) * ODIM + n0 + 16 + lr];
    c0 = __builtin_amdgcn_wmma_f32_16x16x4_f32(false, a, false, b0, (short)0,
                                               c0, false, false);
    c1 = __builtin_amdgcn_wmma_f32_16x16x4_f32(false, a, false, b1, (short)0,
                                               c1, false, false);
  }

  const float bb0 = bias[n0 + lr];
  const float bb1 = bias[n0 + 16 + lr];
#pragma unroll
  for (int r = 0; r < 8; ++r) {
    const int m = m0 + r + 8 * half;
    out[(size_t)m * ODIM + n0 + lr]      = c0[r] + bb0;
    out[(size_t)m * ODIM + n0 + 16 + lr] = c1[r] + bb1;
  }
}

// ---------------------------------------------------------------------------
// Row-wise L2 normalize in place: one wave per 256-float row.
// Lane holds 8 contiguous floats (2 x b128); sum-of-squares via wave32
// shuffle-xor tree; x *= rsqrt(max(sumsq, 1e-12)).
// ---------------------------------------------------------------------------
__global__ __launch_bounds__(256) void l2norm_rows_kernel(float* __restrict__ buf,
                                                          int nrows) {
  const int row  = blockIdx.x * 8 + (threadIdx.x >> 5);
  const int lane = threadIdx.x & 31;
  if (row >= nrows) return;

  v4f* p = (v4f*)(buf + (size_t)row * ODIM);
  v4f x0 = p[lane * 2];
  v4f x1 = p[lane * 2 + 1];
  float s = x0.x * x0.x + x0.y * x0.y + x0.z * x0.z + x0.w * x0.w +
            x1.x * x1.x + x1.y * x1.y + x1.z * x1.z + x1.w * x1.w;
#pragma unroll
  for (int off = 16; off; off >>= 1) s += __shfl_xor(s, off, 32);
  const float inv = rsqrtf(fmaxf(s, 1e-12f));
  x0 *= inv;
  x1 *= inv;
  p[lane * 2]     = x0;
  p[lane * 2 + 1] = x1;
}

// ---------------------------------------------------------------------------
// scores[b,l,m] = (1/16) * sum_d q[b,l,d] * (0.4*k[b,m,d] + 0.6*edge[b,l,m,d])
// One block per (b,l); 8 waves sweep m. Lane holds q[lane*8..lane*8+7] in
// registers. edge is a 512 MiB one-shot stream -> non-temporal loads; k rows
// are reused across 512 blocks -> regular (L2-cached) loads.
// ---------------------------------------------------------------------------
__global__ __launch_bounds__(256) void scores_kernel(
    const float* __restrict__ qn, const float* __restrict__ kn,
    const float* __restrict__ edge, float* __restrict__ scores) {
  const int bl   = blockIdx.x;           // b*SEQL + l
  const int b    = bl >> 9;
  const int wave = threadIdx.x >> 5;
  const int lane = threadIdx.x & 31;

  const v4f* qrow = (const v4f*)(qn + (size_t)bl * ODIM);
  const v4f q0 = qrow[lane * 2];
  const v4f q1 = qrow[lane * 2 + 1];

  const size_t ebase = (size_t)bl * SEQL * ODIM;

  for (int m = wave; m < SEQL; m += 8) {
    const v4f* krow = (const v4f*)(kn + (size_t)(b * SEQL + m) * ODIM);
    const v4f* erow = (const v4f*)(edge + ebase + (size_t)m * ODIM);
    const v4f k0 = krow[lane * 2];
    const v4f k1 = krow[lane * 2 + 1];
    const v4f e0 = __builtin_nontemporal_load(erow + lane * 2);
    const v4f e1 = __builtin_nontemporal_load(erow + lane * 2 + 1);

    float p = q0.x * (0.4f * k0.x + 0.6f * e0.x) +
              q0.y * (0.4f * k0.y + 0.6f * e0.y) +
              q0.z * (0.4f * k0.z + 0.6f * e0.z) +
              q0.w * (0.4f * k0.w + 0.6f * e0.w) +
              q1.x * (0.4f * k1.x + 0.6f * e1.x) +
              q1.y * (0.4f * k1.y + 0.6f * e1.y) +
              q1.z * (0.4f * k1.z + 0.6f * e1.z) +
              q1.w * (0.4f * k1.w + 0.6f * e1.w);
#pragma unroll
    for (int off = 16; off; off >>= 1) p += __shfl_xor(p, off, 32);
    if (lane == 0) scores[(size_t)bl * SEQL + m] = p * 0.0625f;  // rsqrt(256)
  }
}

extern "C" void kernel_launch(void* const* d_in, const int* in_sizes, int n_in,
                              void* d_out, int out_size, void* d_ws,
                              size_t ws_size, hipStream_t stream) {
  (void)in_sizes; (void)n_in; (void)out_size; (void)ws_size;

  const float* queries = (const float*)d_in[0];
  const float* keys    = (const float*)d_in[1];
  const float* values  = (const float*)d_in[2];
  // d_in[3] relative_encoding_lookup: dead code in the reference -> untouched.
  const float* edge = (const float*)d_in[4];
  const float* Wq = (const float*)d_in[5];
  const float* bq = (const float*)d_in[6];
  const float* Wk = (const float*)d_in[7];
  const float* bk = (const float*)d_in[8];
  const float* Wv = (const float*)d_in[9];
  const float* bv = (const float*)d_in[10];

  float* out    = (float*)d_out;
  float* scores = out;                                    // [B,L,L]
  float* v_out  = out + (size_t)BATCH * SEQL * SEQL;      // [B,L,OD]
  float* q_out  = v_out + (size_t)BATCH * SEQL * ODIM;    // [B,L,OD]
  float* k_buf  = (float*)d_ws;                           // [B,L,OD] scratch

  const int M = BATCH * SEQL;  // 1024
  dim3 blk(256);

  proj_wmma_kernel<<<dim3(M / 16), blk, 0, stream>>>(queries, Wq, bq, q_out);
  proj_wmma_kernel<<<dim3(M / 16), blk, 0, stream>>>(keys,    Wk, bk, k_buf);
  proj_wmma_kernel<<<dim3(M / 16), blk, 0, stream>>>(values,  Wv, bv, v_out);

  l2norm_rows_kernel<<<dim3(M / 8), blk, 0, stream>>>(q_out, M);
  l2norm_rows_kernel<<<dim3(M / 8), blk, 0, stream>>>(k_buf, M);
  l2norm_rows_kernel<<<dim3(M / 8), blk, 0, stream>>>(v_out, M);

  scores_kernel<<<dim3(BATCH * SEQL), blk, 0, stream>>>(q_out, k_buf, edge,
                                                        scores);
}